// FinalCoarseToFineDensenSampleIGAModulev2_9182640078988
// MI455X (gfx1250) — compile-verified
//
#include <hip/hip_runtime.h>
#include <hip/hip_bf16.h>
#include <math.h>

typedef __attribute__((ext_vector_type(2))) float v2f;
typedef __attribute__((ext_vector_type(8))) float v8f;

namespace {
constexpr int   B_ = 4, K_ = 512, N_ = 4096, M_ = 4;
constexpr float CHI2_95  = 7.815f;
constexpr float JITTER   = 1e-6f;
constexpr float ALPHA    = 1.5f;
constexpr float SIG_FLOOR = 1e-3f;
constexpr float SIG_CEIL  = 100.0f;
constexpr float BETA     = 0.85f;
constexpr int   KNN_     = 8;
constexpr float BIG      = 1e9f;   // same sentinel as reference
constexpr float HUGE_    = 3.0e38f;
}

// ---------------------------------------------------------------------------
// Kernel 1: largest-remainder quota + inclusive cumsum.  One block per batch.
// ---------------------------------------------------------------------------
__global__ __launch_bounds__(K_) void quota_csum_kernel(
    const float* __restrict__ occ, const float* __restrict__ mask_parent,
    int* __restrict__ csum)
{
  __shared__ float red[K_];
  __shared__ float fr[K_];
  __shared__ int   ib[K_];
  const int b = blockIdx.x;
  const int k = threadIdx.x;

  const float mk    = mask_parent[b * K_ + k];
  const bool  valid = mk > 0.5f;
  const float pi    = valid ? occ[b * K_ + k] : 0.0f;

  red[k] = pi; __syncthreads();
  for (int s = K_ / 2; s > 0; s >>= 1) { if (k < s) red[k] += red[k + s]; __syncthreads(); }
  const float S = fmaxf(red[0], 1e-9f);
  __syncthreads();

  const float raw  = (pi / S) * (float)N_;
  const float base = floorf(raw);

  red[k] = base; __syncthreads();
  for (int s = K_ / 2; s > 0; s >>= 1) { if (k < s) red[k] += red[k + s]; __syncthreads(); }
  const int rem = (int)((float)N_ - red[0]);
  __syncthreads();

  const float frac = valid ? (raw - base) : -1e9f;
  fr[k] = frac; __syncthreads();

  // rank = position in stable descending sort of frac
  int rank = 0;
  for (int j = 0; j < K_; ++j) {
    const float fj = fr[j];
    rank += ((fj > frac) || ((fj == frac) && (j < k))) ? 1 : 0;
  }
  const int nk = valid ? ((int)base + ((rank < rem) ? 1 : 0)) : 0;

  // inclusive scan (Hillis-Steele)
  ib[k] = nk; __syncthreads();
  for (int off = 1; off < K_; off <<= 1) {
    const int self = ib[k];
    const int add  = (k >= off) ? ib[k - off] : 0;
    __syncthreads();
    ib[k] = self + add;
    __syncthreads();
  }
  csum[b * K_ + k] = ib[k];
}

// ---------------------------------------------------------------------------
// Kernel 2: parent assignment (searchsorted) + Cholesky + candidate pick.
// ---------------------------------------------------------------------------
__global__ __launch_bounds__(256) void assign_sample_kernel(
    const float* __restrict__ mu_p, const float* __restrict__ Sig_p,
    const float* __restrict__ node_mask, const float* __restrict__ z,
    const int* __restrict__ csum,
    float* __restrict__ mu0_out, float* __restrict__ parent_out_f,
    int* __restrict__ parent_ws)
{
  const int t = blockIdx.x * blockDim.x + threadIdx.x;
  if (t >= B_ * N_) return;
  const int b = t / N_, n = t % N_;

  // searchsorted(csum, n, side='right') == count of csum[k] <= n
  const int* cs = csum + b * K_;
  int lo = 0, hi = K_;
  while (lo < hi) { const int mid = (lo + hi) >> 1; if (cs[mid] <= n) lo = mid + 1; else hi = mid; }
  const int parent = (lo < (K_ - 1)) ? lo : (K_ - 1);

  const float* mu = mu_p + (size_t)(b * K_ + parent) * 3;
  const float* S  = Sig_p + (size_t)(b * K_ + parent) * 9;

  // Sig_s = sym(Sig_p * SHRINK) + JITTER * I   (SHRINK == 1)
  float s00 = S[0] + JITTER, s11 = S[4] + JITTER, s22 = S[8] + JITTER;
  float s01 = 0.5f * (S[1] + S[3]);
  float s02 = 0.5f * (S[2] + S[6]);
  float s12 = 0.5f * (S[5] + S[7]);

  // manual 3x3 Cholesky, eps = 1e-6, clip below only
  const float eps = 1e-6f;
  const float l00 = sqrtf(fmaxf(s00, eps));
  const float l10 = s01 / fmaxf(l00, eps);
  const float l20 = s02 / fmaxf(l00, eps);
  const float l11 = sqrtf(fmaxf(s11 - l10 * l10, eps));
  const float l21 = (s12 - l20 * l10) / fmaxf(l11, eps);
  const float l22 = sqrtf(fmaxf(s22 - l20 * l20 - l21 * l21, eps));

  // pick first candidate inside chi2 ellipsoid, else min maha2
  const float* zp = z + ((size_t)(b * N_ + n) * M_) * 3;
  int   pick = -1, besti = 0;
  float bestm = HUGE_;
  #pragma unroll
  for (int m = 0; m < M_; ++m) {
    const float zx = zp[m * 3 + 0], zy = zp[m * 3 + 1], zz = zp[m * 3 + 2];
    const float maha = zx * zx + zy * zy + zz * zz;
    if ((maha <= CHI2_95) && (pick < 0)) pick = m;
    if (maha < bestm) { bestm = maha; besti = m; }
  }
  if (pick < 0) pick = besti;

  const float zx = zp[pick * 3 + 0], zy = zp[pick * 3 + 1], zz = zp[pick * 3 + 2];
  const float nm = node_mask[b * N_ + n];
  const float cx = mu[0] + l00 * zx;
  const float cy = mu[1] + l10 * zx + l11 * zy;
  const float cz = mu[2] + l20 * zx + l21 * zy + l22 * zz;

  float* o = mu0_out + (size_t)(b * N_ + n) * 3;
  o[0] = cx * nm; o[1] = cy * nm; o[2] = cz * nm;
  parent_ws[b * N_ + n]    = parent;
  parent_out_f[b * N_ + n] = (float)parent;
}

// ---------------------------------------------------------------------------
// Kernel 3: kNN spacing sigma via V_WMMA_F32_16X16X4_F32 distance tiles.
// One wave32 per 16-row strip; 256 column tiles of 16; per-lane sorted top-8.
// D[i][j] = (-2x_i, -2y_i, -2z_i, 1) . (x_j, y_j, z_j, sq_j);  d2 = D + sq_i
// ---------------------------------------------------------------------------
__global__ __launch_bounds__(32) void knn_sigma_wmma_kernel(
    const float* __restrict__ mu0,        // [B,N,3]
    const float* __restrict__ node_mask,  // [B,N]
    float* __restrict__ sigma_ws)         // [B,N]
{
  __shared__ float rx[16], ry[16], rz[16], rsq_s[16];
  __shared__ int   rv[16];
  __shared__ float mergebuf[16 * 16 * KNN_];   // [localRow][laneCol][KNN]

  const int wave    = blockIdx.x;
  const int b       = wave / (N_ / 16);
  const int strip   = wave % (N_ / 16);
  const int rowBase = strip * 16;
  const int lane    = threadIdx.x;
  const int li      = lane & 15;
  const int half    = lane >> 4;

  if (lane < 16) {
    const int r = rowBase + lane;
    const float x = mu0[(size_t)(b * N_ + r) * 3 + 0];
    const float y = mu0[(size_t)(b * N_ + r) * 3 + 1];
    const float w = mu0[(size_t)(b * N_ + r) * 3 + 2];
    rx[lane] = x; ry[lane] = y; rz[lane] = w;
    rsq_s[lane] = x * x + y * y + w * w;
    rv[lane] = (node_mask[b * N_ + r] > 0.5f) ? 1 : 0;
  }
  __syncthreads();

  // A fragment (16x4 f32): lanes 0-15 hold K=0,1; lanes 16-31 hold K=2,3
  v2f a;
  a[0] = half ? (-2.0f * rz[li]) : (-2.0f * rx[li]);
  a[1] = half ? 1.0f             : (-2.0f * ry[li]);

  float rsq[8]; int rvalid[8];
  #pragma unroll
  for (int v = 0; v < 8; ++v) { rsq[v] = rsq_s[half * 8 + v]; rvalid[v] = rv[half * 8 + v]; }

  float s[8][KNN_];
  #pragma unroll
  for (int v = 0; v < 8; ++v)
    #pragma unroll
    for (int tt = 0; tt < KNN_; ++tt) s[v][tt] = HUGE_;

  for (int jt = 0; jt < N_ / 16; ++jt) {
    const int col = jt * 16 + li;
    const float cx = mu0[(size_t)(b * N_ + col) * 3 + 0];
    const float cy = mu0[(size_t)(b * N_ + col) * 3 + 1];
    const float cz = mu0[(size_t)(b * N_ + col) * 3 + 2];
    const float csq = cx * cx + cy * cy + cz * cz;
    const bool cvalid = node_mask[b * N_ + col] > 0.5f;

    // B fragment (4x16 f32): VGPR0: K=0 (lanes 0-15) / K=1 (lanes 16-31);
    //                        VGPR1: K=2 (lanes 0-15) / K=3 (lanes 16-31)
    v2f bf;
    bf[0] = half ? cy : cx;
    bf[1] = half ? csq : cz;

    v8f c = {0.f, 0.f, 0.f, 0.f, 0.f, 0.f, 0.f, 0.f};
    v8f d = __builtin_amdgcn_wmma_f32_16x16x4_f32(
        /*neg_a=*/false, a, /*neg_b=*/false, bf,
        /*c_mod=*/(short)0, c, /*reuse_a=*/false, /*reuse_b=*/false);

    #pragma unroll
    for (int v = 0; v < 8; ++v) {
      const int row = rowBase + half * 8 + v;
      float val = d[v] + rsq[v];
      const bool bad = (col == row) || (!cvalid) || (!rvalid[v]);
      val = bad ? BIG : val;
      // branchless sorted top-KNN insert: replace max, one bubble pass
      s[v][KNN_ - 1] = fminf(s[v][KNN_ - 1], val);
      #pragma unroll
      for (int tt = KNN_ - 1; tt > 0; --tt) {
        const float mlo = fminf(s[v][tt - 1], s[v][tt]);
        const float mhi = fmaxf(s[v][tt - 1], s[v][tt]);
        s[v][tt - 1] = mlo; s[v][tt] = mhi;
      }
    }
  }

  // dump per-lane top-8 lists, then one lane per row merges 16x8 candidates
  #pragma unroll
  for (int v = 0; v < 8; ++v) {
    const int lr = half * 8 + v;
    #pragma unroll
    for (int tt = 0; tt < KNN_; ++tt)
      mergebuf[(lr * 16 + li) * KNN_ + tt] = s[v][tt];
  }
  __syncthreads();

  if (lane < 16) {
    float* rowv = &mergebuf[lane * 16 * KNN_];
    float acc = 0.0f;
    for (int tt = 0; tt < KNN_; ++tt) {
      float best = HUGE_; int bi = 0;
      for (int q = 0; q < 16 * KNN_; ++q) {
        const float vq = rowv[q];
        if (vq < best) { best = vq; bi = q; }
      }
      rowv[bi] = HUGE_;
      acc += sqrtf(fmaxf(best, 0.0f));
    }
    const float dmean = acc * (1.0f / (float)KNN_);
    const float sg = fminf(fmaxf(ALPHA * dmean, SIG_FLOOR), SIG_CEIL);
    sigma_ws[b * N_ + rowBase + lane] = sg;
  }
}

// ---------------------------------------------------------------------------
// Kernel 4: Sig0 = BETA*sym(Sig_p[parent]) + (1-BETA)*Sig_spacing, + jitter.
// ---------------------------------------------------------------------------
__global__ __launch_bounds__(256) void sigma_blend_kernel(
    const float* __restrict__ Sig_p, const float* __restrict__ node_mask,
    const int* __restrict__ parent_ws, const float* __restrict__ sigma_ws,
    float* __restrict__ Sig0_out)
{
  const int t = blockIdx.x * blockDim.x + threadIdx.x;
  if (t >= B_ * N_) return;
  const int b = t / N_, n = t % N_;

  const int p = parent_ws[b * N_ + n];
  const float* S = Sig_p + (size_t)(b * K_ + p) * 9;
  const float s00 = S[0] + JITTER, s11 = S[4] + JITTER, s22 = S[8] + JITTER;
  const float s01 = 0.5f * (S[1] + S[3]);
  const float s02 = 0.5f * (S[2] + S[6]);
  const float s12 = 0.5f * (S[5] + S[7]);

  const float sg = sigma_ws[b * N_ + n];
  const float sp = sg * sg + JITTER;   // Sig_spacing diagonal

  const float o00 = BETA * s00 + (1.0f - BETA) * sp + JITTER;
  const float o11 = BETA * s11 + (1.0f - BETA) * sp + JITTER;
  const float o22 = BETA * s22 + (1.0f - BETA) * sp + JITTER;
  const float o01 = BETA * s01, o02 = BETA * s02, o12 = BETA * s12;

  const float nm = node_mask[b * N_ + n];
  float* o = Sig0_out + (size_t)(b * N_ + n) * 9;
  o[0] = o00 * nm; o[1] = o01 * nm; o[2] = o02 * nm;
  o[3] = o01 * nm; o[4] = o11 * nm; o[5] = o12 * nm;
  o[6] = o02 * nm; o[7] = o12 * nm; o[8] = o22 * nm;
}

// ---------------------------------------------------------------------------
extern "C" void kernel_launch(void* const* d_in, const int* in_sizes, int n_in,
                              void* d_out, int out_size, void* d_ws, size_t ws_size,
                              hipStream_t stream) {
  (void)in_sizes; (void)n_in; (void)out_size; (void)ws_size;
  const float* mu_p        = (const float*)d_in[0];
  const float* Sig_p       = (const float*)d_in[1];
  const float* occ         = (const float*)d_in[2];
  const float* mask_parent = (const float*)d_in[3];
  const float* node_mask   = (const float*)d_in[4];
  const float* z           = (const float*)d_in[5];

  float* out      = (float*)d_out;
  float* mu0      = out;                              // [B,N,3]
  float* Sig0     = out + (size_t)B_ * N_ * 3;        // [B,N,3,3]
  float* parent_f = out + (size_t)B_ * N_ * 12;       // [B,N] as float

  char* ws = (char*)d_ws;
  int*   csum      = (int*)ws;                                        // B*K
  int*   parent_ws = (int*)(ws + sizeof(int) * (size_t)B_ * K_);      // B*N
  float* sigma_ws  = (float*)(ws + sizeof(int) * ((size_t)B_ * K_ + (size_t)B_ * N_)); // B*N

  quota_csum_kernel<<<B_, K_, 0, stream>>>(occ, mask_parent, csum);
  assign_sample_kernel<<<(B_ * N_ + 255) / 256, 256, 0, stream>>>(
      mu_p, Sig_p, node_mask, z, csum, mu0, parent_f, parent_ws);
  knn_sigma_wmma_kernel<<<B_ * (N_ / 16), 32, 0, stream>>>(mu0, node_mask, sigma_ws);
  sigma_blend_kernel<<<(B_ * N_ + 255) / 256, 256, 0, stream>>>(
      Sig_p, node_mask, parent_ws, sigma_ws, Sig0);
}